// ReciprocatorBlock_18245021073727
// MI455X (gfx1250) — compile-verified
//
#include <hip/hip_runtime.h>
#include <math.h>

// ---------------------------------------------------------------------------
// ReciprocatorBlock on MI455X (gfx1250, wave32, WMMA)
//   B=8, S=512, H=512, D1=D2=64, N=4096, INNER=1024
// ---------------------------------------------------------------------------

typedef __attribute__((ext_vector_type(2))) float v2f;
typedef __attribute__((ext_vector_type(8))) float v8f;

#define B_      8
#define S_      512
#define H_      512
#define N_      4096
#define INNER_  1024
#define ROWS_   (B_ * S_)        // 4096 rows (B*S)
#define EPSF    1e-6f

// ---------------------------------------------------------------------------
// Constant setup: DFT matrices F (fwd) / Fi (inv, conj/64), spectral gain G,
// decay lam = exp(-softplus(decay_log) + i*decay_phase).  idx in [0,4096).
// ---------------------------------------------------------------------------
__global__ void setup_consts(const float* __restrict__ decay_log,
                             const float* __restrict__ decay_phase,
                             float* __restrict__ Fre,  float* __restrict__ Fim,
                             float* __restrict__ FiRe, float* __restrict__ FiIm,
                             float* __restrict__ G,
                             float* __restrict__ lamRe, float* __restrict__ lamIm)
{
    const int idx = blockIdx.x * blockDim.x + threadIdx.x;
    if (idx >= 4096) return;
    const int j = idx >> 6, k = idx & 63;

    // DFT: F[j,k] = exp(-2*pi*i*j*k/64); periodic in (j*k) mod 64
    const int p = (j * k) & 63;
    const float theta = (float)p * (6.28318530717958647692f / 64.0f);
    float sn, cs;
    sincosf(theta, &sn, &cs);
    Fre[idx]  = cs;
    Fim[idx]  = -sn;
    FiRe[idx] = cs * (1.0f / 64.0f);
    FiIm[idx] = sn * (1.0f / 64.0f);

    // Spectral gain (fftfreq grid)
    const float fy = (float)(j < 32 ? j : j - 64) * (1.0f / 64.0f);
    const float fx = (float)(k < 32 ? k : k - 64) * (1.0f / 64.0f);
    const float r  = sqrtf(fy * fy + fx * fx);
    G[idx] = 0.5f * expf(-(r * r) / (2.0f * 0.35f * 0.35f))
           + 0.5f * ((r >= 0.5f) ? 1.0f : 0.0f);

    // lam = exp(-softplus(decay_log)) * exp(i*phase)
    const float dl  = decay_log[idx];
    const float sp  = (dl > 20.0f) ? dl : log1pf(expf(dl));
    const float mag = expf(-sp);
    float sph, cph;
    sincosf(decay_phase[idx], &sph, &cph);
    lamRe[idx] = mag * cph;
    lamIm[idx] = mag * sph;
}

// ---------------------------------------------------------------------------
// Complex GEMM with fp32 WMMA (V_WMMA_F32_16X16X4_F32).
//   C = A(MxK, split re/im planes) * W(KxN, complex interleaved) + bias
// Block: 128 threads = 4 waves, 32x32 output tile, each wave one 16x16 tile.
// 4 fp32 accumulators per wave give the complex product:
//   Cre = Ar*Br - Ai*Bi ; Cim = Ar*Bi + Ai*Br
// EPI: 0 = write planes, 1 = modReLU then planes, 2 = residual + interleave
// ---------------------------------------------------------------------------
template <int EPI>
__global__ __launch_bounds__(128)
void cgemm_wmma(const float* __restrict__ Are, const float* __restrict__ Aim,
                const float* __restrict__ Wc,  const float* __restrict__ bias,
                float* __restrict__ outRe, float* __restrict__ outIm,
                const float* __restrict__ actb,
                const float* __restrict__ hidRe, const float* __restrict__ hidIm,
                float* __restrict__ outC,
                int M, int K, int N)
{
    __shared__ float As_re[32][33];
    __shared__ float As_im[32][33];
    __shared__ float Bs_re[32][33];
    __shared__ float Bs_im[32][33];

    const int tid  = threadIdx.x;
    const int lane = tid & 31;
    const int wave = tid >> 5;
    const int wm   = wave >> 1;          // 0..1 (M sub-tile)
    const int wn   = wave & 1;           // 0..1 (N sub-tile)
    const int m0   = blockIdx.y * 32;
    const int n0   = blockIdx.x * 32;

    v8f rr = {}, ii = {}, ri = {}, ir = {};

    // WMMA fragment coordinates (ISA 7.12.2, 32-bit A 16x4 / B 4x16)
    const int am = wm * 16 + (lane & 15);      // A row for this lane
    const int bn = wn * 16 + (lane & 15);      // B col for this lane
    const int kh = (lane >> 4) * 2;            // K sub-offset per half-wave

    for (int k0 = 0; k0 < K; k0 += 32) {
        // cooperative LDS staging (dims are all multiples of 32 -> no guards)
        for (int i = tid; i < 32 * 32; i += 128) {
            const int r = i >> 5, c = i & 31;
            As_re[r][c] = Are[(size_t)(m0 + r) * K + (k0 + c)];
            As_im[r][c] = Aim[(size_t)(m0 + r) * K + (k0 + c)];
            const float* wp = Wc + 2u * ((size_t)(k0 + r) * N + (n0 + c));
            Bs_re[r][c] = wp[0];
            Bs_im[r][c] = wp[1];
        }
        __syncthreads();

#pragma unroll
        for (int kk = 0; kk < 32; kk += 4) {
            v2f ar, ai, br, bi;
            ar[0] = As_re[am][kk + kh];     ar[1] = As_re[am][kk + kh + 1];
            ai[0] = As_im[am][kk + kh];     ai[1] = As_im[am][kk + kh + 1];
            br[0] = Bs_re[kk + kh][bn];     br[1] = Bs_re[kk + kh + 1][bn];
            bi[0] = Bs_im[kk + kh][bn];     bi[1] = Bs_im[kk + kh + 1][bn];
            rr = __builtin_amdgcn_wmma_f32_16x16x4_f32(false, ar, false, br, (short)0, rr, false, false);
            ii = __builtin_amdgcn_wmma_f32_16x16x4_f32(false, ai, false, bi, (short)0, ii, false, false);
            ri = __builtin_amdgcn_wmma_f32_16x16x4_f32(false, ar, false, bi, (short)0, ri, false, false);
            ir = __builtin_amdgcn_wmma_f32_16x16x4_f32(false, ai, false, br, (short)0, ir, false, false);
        }
        __syncthreads();
    }

    // Epilogue. C/D layout: VGPR v -> M = v + 8*(lane/16), N = lane%16.
#pragma unroll
    for (int v = 0; v < 8; ++v) {
        const int gm = m0 + wm * 16 + v + ((lane >> 4) << 3);
        const int gn = n0 + wn * 16 + (lane & 15);
        float cre = rr[v] - ii[v] + bias[2 * gn];
        float cim = ri[v] + ir[v] + bias[2 * gn + 1];
        const size_t o = (size_t)gm * N + gn;
        if (EPI == 0) {
            outRe[o] = cre;
            outIm[o] = cim;
        } else if (EPI == 1) {
            const float m = sqrtf(cre * cre + cim * cim);
            const float s = fmaxf(m + actb[gn], 0.0f) / (m + EPSF);
            outRe[o] = s * cre;
            outIm[o] = s * cim;
        } else {
            outC[2 * o]     = hidRe[o] + cre;
            outC[2 * o + 1] = hidIm[o] + cim;
        }
    }
}

// ---------------------------------------------------------------------------
// Spectral filter: per (b,s) item (64x64 complex grid U):
//   U <- Fi * (G .* (F * U * F^T)) * Fi^T   (F symmetric, Fi = conj(F)/64)
// One 256-thread block per item; grids live in LDS; DFT matrices from L2.
// ---------------------------------------------------------------------------
__device__ __forceinline__
void cstage_left(const float* __restrict__ Mre, const float* __restrict__ Mim,
                 const float* Xr, const float* Xi, float* Yr, float* Yi, int tid)
{
    // Y[r,c] = sum_k M[r,k] * X[k,c]
    for (int jj = 0; jj < 16; ++jj) {
        const int o = tid + jj * 256;
        const int r = o >> 6, c = o & 63;
        float accr = 0.0f, acci = 0.0f;
        for (int k = 0; k < 64; ++k) {
            const float mr = Mre[r * 64 + k], mi = Mim[r * 64 + k]; // wave-uniform
            const float xr = Xr[k * 64 + c],  xi = Xi[k * 64 + c];  // conflict-free
            accr += mr * xr - mi * xi;
            acci += mr * xi + mi * xr;
        }
        Yr[o] = accr;
        Yi[o] = acci;
    }
}

__device__ __forceinline__
void cstage_rightT(const float* __restrict__ Mre, const float* __restrict__ Mim,
                   const float* Xr, const float* Xi, float* Yr, float* Yi,
                   const float* __restrict__ gain, int tid)
{
    // Y[r,c] = sum_k X[r,k] * M[c,k]   (optionally * gain[r,c])
    for (int jj = 0; jj < 16; ++jj) {
        const int o = tid + jj * 256;
        const int r = o >> 6, c = o & 63;
        float accr = 0.0f, acci = 0.0f;
        for (int k = 0; k < 64; ++k) {
            const float mr = Mre[c * 64 + k], mi = Mim[c * 64 + k];
            const float xr = Xr[r * 64 + k],  xi = Xi[r * 64 + k];   // broadcast
            accr += xr * mr - xi * mi;
            acci += xr * mi + xi * mr;
        }
        if (gain) { const float g = gain[o]; accr *= g; acci *= g; }
        Yr[o] = accr;
        Yi[o] = acci;
    }
}

__global__ __launch_bounds__(256)
void spectral_filter(float* __restrict__ uRe, float* __restrict__ uIm,
                     const float* __restrict__ Fre,  const float* __restrict__ Fim,
                     const float* __restrict__ FiRe, const float* __restrict__ FiIm,
                     const float* __restrict__ G)
{
    __shared__ float Ar[4096], Ai[4096], Br[4096], Bi[4096];   // 64 KB of 320 KB
    const int tid = threadIdx.x;
    const size_t base = (size_t)blockIdx.x * 4096;

    for (int i = tid; i < 4096; i += 256) { Ar[i] = uRe[base + i]; Ai[i] = uIm[base + i]; }
    __syncthreads();
    cstage_left (Fre,  Fim,  Ar, Ai, Br, Bi, tid);            __syncthreads();
    cstage_rightT(Fre, Fim,  Br, Bi, Ar, Ai, G, tid);         __syncthreads();
    cstage_left (FiRe, FiIm, Ar, Ai, Br, Bi, tid);            __syncthreads();
    // last stage writes straight to global
    for (int jj = 0; jj < 16; ++jj) {
        const int o = tid + jj * 256;
        const int r = o >> 6, c = o & 63;
        float accr = 0.0f, acci = 0.0f;
        for (int k = 0; k < 64; ++k) {
            const float mr = FiRe[c * 64 + k], mi = FiIm[c * 64 + k];
            const float xr = Br[r * 64 + k],   xi = Bi[r * 64 + k];
            accr += xr * mr - xi * mi;
            acci += xr * mi + xi * mr;
        }
        uRe[base + o] = accr;
        uIm[base + o] = acci;
    }
}

// ---------------------------------------------------------------------------
// Sequential scan with per-step RMS normalization.  One block per batch (8),
// 1024 threads, 4 complex state elements per thread held in registers.
// States overwrite u in place; final state written interleaved to d_out tail.
// ---------------------------------------------------------------------------
__global__ __launch_bounds__(1024)
void scan_norm(float* __restrict__ uRe, float* __restrict__ uIm,
               const float* __restrict__ lamRe, const float* __restrict__ lamIm,
               float* __restrict__ finalC)
{
    __shared__ float red[32];
    __shared__ float tot;
    const int b   = blockIdx.x;
    const int tid = threadIdx.x;

    float sre[4], sim[4], lr[4], li[4];
#pragma unroll
    for (int j = 0; j < 4; ++j) {
        const int e = tid + j * 1024;
        sre[j] = 0.0f; sim[j] = 0.0f;
        lr[j] = lamRe[e]; li[j] = lamIm[e];
    }

    for (int t = 0; t < S_; ++t) {
        const size_t rb = ((size_t)b * S_ + t) * (size_t)N_;
        float local = 0.0f;
#pragma unroll
        for (int j = 0; j < 4; ++j) {
            const int e = tid + j * 1024;
            const float ur = uRe[rb + e], ui = uIm[rb + e];
            const float nr = lr[j] * sre[j] - li[j] * sim[j] + ur;
            const float ni = lr[j] * sim[j] + li[j] * sre[j] + ui;
            sre[j] = nr; sim[j] = ni;
            local += nr * nr + ni * ni;
        }
        // wave32 reduce, then cross-wave via LDS
#pragma unroll
        for (int off = 16; off > 0; off >>= 1) local += __shfl_down(local, off, 32);
        if ((tid & 31) == 0) red[tid >> 5] = local;
        __syncthreads();
        if (tid < 32) {
            float v = red[tid];
#pragma unroll
            for (int off = 16; off > 0; off >>= 1) v += __shfl_down(v, off, 32);
            if (tid == 0) tot = v;
        }
        __syncthreads();
        const float n  = sqrtf(tot * (1.0f / 4096.0f));
        const float sc = 1.0f / (n + EPSF);
#pragma unroll
        for (int j = 0; j < 4; ++j) {
            const int e = tid + j * 1024;
            sre[j] *= sc; sim[j] *= sc;
            uRe[rb + e] = sre[j];
            uIm[rb + e] = sim[j];
        }
        __syncthreads();
    }

#pragma unroll
    for (int j = 0; j < 4; ++j) {
        const int e = tid + j * 1024;
        const size_t o = (size_t)b * 4096 + e;
        finalC[2 * o]     = sre[j];
        finalC[2 * o + 1] = sim[j];
    }
}

// ---------------------------------------------------------------------------
// Complex LayerNorm over H=512, in place on re/im planes. One block per row.
// ---------------------------------------------------------------------------
__device__ __forceinline__ float blk_reduce_256(float v, float* sm, int tid)
{
    __syncthreads();
#pragma unroll
    for (int off = 16; off > 0; off >>= 1) v += __shfl_down(v, off, 32);
    if ((tid & 31) == 0) sm[tid >> 5] = v;
    __syncthreads();
    if (tid == 0) {
        float t = 0.0f;
        for (int i = 0; i < 8; ++i) t += sm[i];
        sm[8] = t;
    }
    __syncthreads();
    return sm[8];
}

__global__ __launch_bounds__(256)
void complex_layernorm(float* __restrict__ Xr, float* __restrict__ Xi,
                       const float* __restrict__ gamma, const float* __restrict__ beta)
{
    __shared__ float sm[9];
    const int tid = threadIdx.x;
    const size_t base = (size_t)blockIdx.x * H_;

    float vr[2], vi[2];
    vr[0] = Xr[base + tid];        vi[0] = Xi[base + tid];
    vr[1] = Xr[base + tid + 256];  vi[1] = Xi[base + tid + 256];

    const float muR = blk_reduce_256(vr[0] + vr[1], sm, tid) * (1.0f / H_);
    const float muI = blk_reduce_256(vi[0] + vi[1], sm, tid) * (1.0f / H_);

    const float z0r = vr[0] - muR, z0i = vi[0] - muI;
    const float z1r = vr[1] - muR, z1i = vi[1] - muI;

    const float var = blk_reduce_256(z0r * z0r + z0i * z0i + z1r * z1r + z1i * z1i,
                                     sm, tid) * (1.0f / H_);
    const float inv = rsqrtf(var + 1e-5f);

    {
        const int c = tid;
        const float gr = gamma[2 * c], gi = gamma[2 * c + 1];
        const float ar = z0r * inv,    ai = z0i * inv;
        Xr[base + c] = gr * ar - gi * ai + beta[2 * c];
        Xi[base + c] = gr * ai + gi * ar + beta[2 * c + 1];
    }
    {
        const int c = tid + 256;
        const float gr = gamma[2 * c], gi = gamma[2 * c + 1];
        const float ar = z1r * inv,    ai = z1i * inv;
        Xr[base + c] = gr * ar - gi * ai + beta[2 * c];
        Xi[base + c] = gr * ai + gi * ar + beta[2 * c + 1];
    }
}

// ---------------------------------------------------------------------------
// Launch: full pipeline on `stream`.  All scratch in d_ws (~185 MB floats).
// d_out layout: out (B,S,H) complex interleaved, then final_state (B,64,64).
// ---------------------------------------------------------------------------
extern "C" void kernel_launch(void* const* d_in, const int* in_sizes, int n_in,
                              void* d_out, int out_size, void* d_ws, size_t ws_size,
                              hipStream_t stream)
{
    (void)in_sizes; (void)n_in; (void)out_size; (void)ws_size;

    const float* hidden_re   = (const float*)d_in[0];
    const float* hidden_im   = (const float*)d_in[1];
    const float* W_in        = (const float*)d_in[2];   // (H,N) complex interleaved
    const float* b_in        = (const float*)d_in[3];   // (N,) complex
    const float* decay_log   = (const float*)d_in[4];
    const float* decay_phase = (const float*)d_in[5];
    const float* W_out       = (const float*)d_in[6];   // (N,H) complex
    const float* b_out       = (const float*)d_in[7];
    const float* ln_gamma    = (const float*)d_in[8];
    const float* ln_beta     = (const float*)d_in[9];
    const float* ffn_W1      = (const float*)d_in[10];  // (H,INNER) complex
    const float* ffn_b1      = (const float*)d_in[11];
    const float* ffn_act_b   = (const float*)d_in[12];
    const float* ffn_W2      = (const float*)d_in[13];  // (INNER,H) complex
    const float* ffn_b2      = (const float*)d_in[14];

    float* out = (float*)d_out;

    // Workspace carve-up (floats)
    float* ws = (float*)d_ws;
    const size_t UP = (size_t)ROWS_ * N_;       // 16.78M  (u / states planes)
    const size_t XP = (size_t)ROWS_ * H_;       //  2.10M  (x planes)
    const size_t HP = (size_t)ROWS_ * INNER_;   //  4.19M  (h1 planes)
    float* uRe   = ws;
    float* uIm   = uRe + UP;
    float* xRe   = uIm + UP;
    float* xIm   = xRe + XP;
    float* h1Re  = xIm + XP;
    float* h1Im  = h1Re + HP;
    float* Fre   = h1Im + HP;
    float* Fim   = Fre + 4096;
    float* FiRe  = Fim + 4096;
    float* FiIm  = FiRe + 4096;
    float* G     = FiIm + 4096;
    float* lamRe = G + 4096;
    float* lamIm = lamRe + 4096;

    // 0) constants (DFT matrices, gain, lam)
    setup_consts<<<16, 256, 0, stream>>>(decay_log, decay_phase,
                                         Fre, Fim, FiRe, FiIm, G, lamRe, lamIm);

    // 1) u = hidden @ W_in + b_in        (M=4096, K=512, N=4096)
    cgemm_wmma<0><<<dim3(N_ / 32, ROWS_ / 32), 128, 0, stream>>>(
        hidden_re, hidden_im, W_in, b_in, uRe, uIm,
        nullptr, nullptr, nullptr, nullptr, ROWS_, H_, N_);

    // 2) u <- ifft2(fft2(u) * G)  via dense DFT stages
    spectral_filter<<<ROWS_, 256, 0, stream>>>(uRe, uIm, Fre, Fim, FiRe, FiIm, G);

    // 3) normalized scan (states overwrite u; final state -> d_out tail)
    scan_norm<<<B_, 1024, 0, stream>>>(uRe, uIm, lamRe, lamIm, out + 2 * XP);

    // 4) delta = states @ W_out + b_out  (M=4096, K=4096, N=512)
    cgemm_wmma<0><<<dim3(H_ / 32, ROWS_ / 32), 128, 0, stream>>>(
        uRe, uIm, W_out, b_out, xRe, xIm,
        nullptr, nullptr, nullptr, nullptr, ROWS_, N_, H_);

    // 5) complex layernorm (in place)
    complex_layernorm<<<ROWS_, 256, 0, stream>>>(xRe, xIm, ln_gamma, ln_beta);

    // 6) h1 = modReLU(x @ W1 + b1)       (M=4096, K=512, N=1024)
    cgemm_wmma<1><<<dim3(INNER_ / 32, ROWS_ / 32), 128, 0, stream>>>(
        xRe, xIm, ffn_W1, ffn_b1, h1Re, h1Im,
        ffn_act_b, nullptr, nullptr, nullptr, ROWS_, H_, INNER_);

    // 7) out = hidden + h1 @ W2 + b2     (M=4096, K=1024, N=512), interleaved
    cgemm_wmma<2><<<dim3(H_ / 32, ROWS_ / 32), 128, 0, stream>>>(
        h1Re, h1Im, ffn_W2, ffn_b2, nullptr, nullptr,
        nullptr, hidden_re, hidden_im, out, ROWS_, INNER_, H_);
}